// Quanvolution2D_mio_28759101014684
// MI455X (gfx1250) — compile-verified
//
#include <hip/hip_runtime.h>
#include <math.h>

typedef __attribute__((ext_vector_type(2))) float v2f;
typedef __attribute__((ext_vector_type(8))) float v8f;

// Problem constants (from reference)
static constexpr int BS = 8, CH = 3, H = 40, W = 40;
static constexpr int HO = 39, WO = 39;
static constexpr int POS_PER_IMG = HO * WO;          // 1521 positions per image
static constexpr int NPOS = BS * POS_PER_IMG;        // 12168 (channels handled inside)
static constexpr int NTILES = (NPOS + 15) / 16;      // 761

// ---------------------------------------------------------------------------
// Kernel 0: build M_q = V_s^H D_q V_s for q = 0,1 from the entangler weights.
// Simulates the 16 embedding-basis columns (|j> placed at index j<<4) through
// 2 layers of (RX on every qubit, CNOT ring). Qubit q lives at bit (7-q).
// Output (d_ws floats): [0..255]=A0 (Re M0), [256..511]=B0 (Im M0),
//                       [512..767]=A1,       [768..1023]=B1, row-major [j][k].
// ---------------------------------------------------------------------------
__global__ __launch_bounds__(256) void build_mats_kernel(
    const float* __restrict__ wts, float* __restrict__ mats) {
  __shared__ float sre[16][256];
  __shared__ float sim2[16][256];
  const int x = threadIdx.x;  // amplitude index 0..255

#pragma unroll
  for (int j = 0; j < 16; ++j) {
    sre[j][x] = (x == (j << 4)) ? 1.0f : 0.0f;
    sim2[j][x] = 0.0f;
  }
  __syncthreads();

  for (int l = 0; l < 2; ++l) {
    // RX(w[l][q]) on each qubit: a'(x) = c*a(x) - i*s*a(x^mask)
    for (int q = 0; q < 8; ++q) {
      float wv = wts[l * 8 + q];
      float s, c;
      sincosf(0.5f * wv, &s, &c);
      const int p = x ^ (1 << (7 - q));
      float nr[16], ni[16];
#pragma unroll
      for (int j = 0; j < 16; ++j) {
        float r1 = sre[j][x], i1 = sim2[j][x];
        float r2 = sre[j][p], i2 = sim2[j][p];
        nr[j] = c * r1 + s * i2;   // Re(c*a - i*s*b)
        ni[j] = c * i1 - s * r2;   // Im(c*a - i*s*b)
      }
      __syncthreads();
#pragma unroll
      for (int j = 0; j < 16; ++j) { sre[j][x] = nr[j]; sim2[j][x] = ni[j]; }
      __syncthreads();
    }
    // CNOT ring: CNOT(q, (q+1)%8); a'(x) = a(x with tgt bit flipped if ctrl set)
    for (int q = 0; q < 8; ++q) {
      const int cm = 1 << (7 - q);
      const int tm = 1 << (7 - ((q + 1) & 7));
      const int xp = (x & cm) ? (x ^ tm) : x;
      float nr[16], ni[16];
#pragma unroll
      for (int j = 0; j < 16; ++j) { nr[j] = sre[j][xp]; ni[j] = sim2[j][xp]; }
      __syncthreads();
#pragma unroll
      for (int j = 0; j < 16; ++j) { sre[j][x] = nr[j]; sim2[j][x] = ni[j]; }
      __syncthreads();
    }
  }

  // M_q[j][k] = sum_x d_q(x) * conj(v_j(x)) * v_k(x);  d_q from bit (7-q) of x
  const int j = x >> 4, k = x & 15;
  float ar0 = 0.f, ai0 = 0.f, ar1 = 0.f, ai1 = 0.f;
  for (int t = 0; t < 256; ++t) {
    float vjr = sre[j][t], vji = sim2[j][t];
    float vkr = sre[k][t], vki = sim2[k][t];
    float re = vjr * vkr + vji * vki;
    float im = vjr * vki - vji * vkr;
    float d0 = ((t >> 7) & 1) ? -1.f : 1.f;
    float d1 = ((t >> 6) & 1) ? -1.f : 1.f;
    ar0 += d0 * re; ai0 += d0 * im;
    ar1 += d1 * re; ai1 += d1 * im;
  }
  mats[0 * 256 + x] = ar0;
  mats[1 * 256 + x] = ai0;
  mats[2 * 256 + x] = ar1;
  mats[3 * 256 + x] = ai1;
}

// ---------------------------------------------------------------------------
// Main kernel: one wave (32 lanes) per block, 16 output positions per wave.
// Per channel: build u,v (embedding real/imag parts) per lane, run
//   G_uA = Pu*A, G_vA = Pv*A, G_uB = Pu*B  per filter via v_wmma_f32_16x16x4,
// finish z = rowdot(G_uA,Pu) + rowdot(G_vA,Pv) - 2*rowdot(G_uB,Pv),
// accumulate over 3 channels, reduce across lanes, write mean to d_out.
// ---------------------------------------------------------------------------
__global__ __launch_bounds__(32) void quanv_kernel(
    const float* __restrict__ xin, const float* __restrict__ mats,
    float* __restrict__ out) {
  const int lane = threadIdx.x;
  const int m = lane & 15;   // patch row (A-operand M index, also B/C/D column N)
  const int h = lane >> 4;   // lane half selects K pair / C-row offset
  const int tile = blockIdx.x;

  __shared__ float Pu[16][16];
  __shared__ float Pv[16][16];

  // Preload B operands: Mat rows (4kb+2h, 4kb+2h+1), column n = m.
  v2f bA0[4], bB0[4], bA1[4], bB1[4];
#pragma unroll
  for (int kb = 0; kb < 4; ++kb) {
    const int r0 = 4 * kb + 2 * h;
    bA0[kb] = (v2f){mats[0 * 256 + r0 * 16 + m], mats[0 * 256 + (r0 + 1) * 16 + m]};
    bB0[kb] = (v2f){mats[1 * 256 + r0 * 16 + m], mats[1 * 256 + (r0 + 1) * 16 + m]};
    bA1[kb] = (v2f){mats[2 * 256 + r0 * 16 + m], mats[2 * 256 + (r0 + 1) * 16 + m]};
    bB1[kb] = (v2f){mats[3 * 256 + r0 * 16 + m], mats[3 * 256 + (r0 + 1) * 16 + m]};
  }

  int pos = tile * 16 + m;
  int posc = (pos < NPOS) ? pos : (NPOS - 1);   // clamp for safe loads
  int b = posc / POS_PER_IMG;
  int rem = posc - b * POS_PER_IMG;
  int ii = rem / WO;
  int jj = rem - ii * WO;

  float z0[8], z1[8];
#pragma unroll
  for (int r = 0; r < 8; ++r) { z0[r] = 0.f; z1[r] = 0.f; }

  for (int ch = 0; ch < 3; ++ch) {
    const float* img = xin + (((size_t)b * CH + ch) * H + ii) * W + jj;
    float t0 = img[0], t1 = img[1], t2 = img[W], t3 = img[W + 1];
    float c0, s0, c1, s1, c2, s2, c3, s3;
    sincosf(0.5f * t0, &s0, &c0);
    sincosf(0.5f * t1, &s1, &c1);
    sincosf(0.5f * t2, &s2, &c2);
    sincosf(0.5f * t3, &s3, &c3);

    // Embedding: psi(e) = prod_t (c_t or -i s_t); phase (-i)^popcount(e)
    float u[16], v[16];
#pragma unroll
    for (int e = 0; e < 16; ++e) {
      float m0 = (e & 8) ? s0 : c0;  // qubit 0 <-> bit 3 of e
      float m1 = (e & 4) ? s1 : c1;
      float m2 = (e & 2) ? s2 : c2;
      float m3 = (e & 1) ? s3 : c3;
      float mag = m0 * m1 * m2 * m3;
      int p = __popc(e) & 3;
      u[e] = (p == 0) ? mag : ((p == 2) ? -mag : 0.0f);
      v[e] = (p == 1) ? -mag : ((p == 3) ? mag : 0.0f);
    }
    if (h == 0) {
#pragma unroll
      for (int e = 0; e < 16; ++e) { Pu[m][e] = u[e]; Pv[m][e] = v[e]; }
    }
    __syncthreads();

    v8f gA0u = {0, 0, 0, 0, 0, 0, 0, 0};
    v8f gA0v = {0, 0, 0, 0, 0, 0, 0, 0};
    v8f gB0u = {0, 0, 0, 0, 0, 0, 0, 0};
    v8f gA1u = {0, 0, 0, 0, 0, 0, 0, 0};
    v8f gA1v = {0, 0, 0, 0, 0, 0, 0, 0};
    v8f gB1u = {0, 0, 0, 0, 0, 0, 0, 0};
#pragma unroll
    for (int kb = 0; kb < 4; ++kb) {
      v2f au = (v2f){u[4 * kb + 2 * h], u[4 * kb + 2 * h + 1]};
      v2f av = (v2f){v[4 * kb + 2 * h], v[4 * kb + 2 * h + 1]};
      gA0u = __builtin_amdgcn_wmma_f32_16x16x4_f32(false, au, false, bA0[kb],
                                                   (short)0, gA0u, false, false);
      gA0v = __builtin_amdgcn_wmma_f32_16x16x4_f32(false, av, false, bA0[kb],
                                                   (short)0, gA0v, false, false);
      gB0u = __builtin_amdgcn_wmma_f32_16x16x4_f32(false, au, false, bB0[kb],
                                                   (short)0, gB0u, false, false);
      gA1u = __builtin_amdgcn_wmma_f32_16x16x4_f32(false, au, false, bA1[kb],
                                                   (short)0, gA1u, false, false);
      gA1v = __builtin_amdgcn_wmma_f32_16x16x4_f32(false, av, false, bA1[kb],
                                                   (short)0, gA1v, false, false);
      gB1u = __builtin_amdgcn_wmma_f32_16x16x4_f32(false, au, false, bB1[kb],
                                                   (short)0, gB1u, false, false);
    }

    // Row-dot finish: lane holds D[m=r+8h][n=lane&15] in vgpr r.
#pragma unroll
    for (int r = 0; r < 8; ++r) {
      float pu = Pu[r + 8 * h][m];
      float pv = Pv[r + 8 * h][m];
      z0[r] += gA0u[r] * pu + (gA0v[r] - 2.0f * gB0u[r]) * pv;
      z1[r] += gA1u[r] * pu + (gA1v[r] - 2.0f * gB1u[r]) * pv;
    }
    __syncthreads();
  }

  // Reduce over the 16 columns (lanes within each half share the row set)
#pragma unroll
  for (int r = 0; r < 8; ++r) {
    float a = z0[r], c = z1[r];
    for (int mask = 1; mask <= 8; mask <<= 1) {
      a += __shfl_xor(a, mask, 32);
      c += __shfl_xor(c, mask, 32);
    }
    z0[r] = a; z1[r] = c;
  }

  if (m == 0) {
#pragma unroll
    for (int r = 0; r < 8; ++r) {
      int posm = tile * 16 + r + 8 * h;
      if (posm < NPOS) {
        int bb = posm / POS_PER_IMG;
        int rr = posm - bb * POS_PER_IMG;
        out[(bb * 2 + 0) * POS_PER_IMG + rr] = z0[r] * (1.0f / 3.0f);
        out[(bb * 2 + 1) * POS_PER_IMG + rr] = z1[r] * (1.0f / 3.0f);
      }
    }
  }
}

extern "C" void kernel_launch(void* const* d_in, const int* in_sizes, int n_in,
                              void* d_out, int out_size, void* d_ws, size_t ws_size,
                              hipStream_t stream) {
  const float* x = (const float*)d_in[0];      // (8,3,40,40) f32
  const float* wts = (const float*)d_in[1];    // (2,8) f32
  float* out = (float*)d_out;                  // (8,2,39,39) f32
  float* mats = (float*)d_ws;                  // 1024 floats

  build_mats_kernel<<<1, 256, 0, stream>>>(wts, mats);
  quanv_kernel<<<NTILES, 32, 0, stream>>>(x, mats, out);
}